// Model_37168646980094
// MI455X (gfx1250) — compile-verified
//
#include <hip/hip_runtime.h>
#include <hip/hip_bf16.h>

// Problem sizes
constexpr int NV = 97;     // vocab
constexpr int ND = 256;    // embed
constexpr int NH = 1024;   // hidden
constexpr int NL = 1024;   // sequence length
constexpr int NG = 4 * NH; // 4096 gate rows per layer
constexpr int NWG = 32;    // persistent workgroups (must be co-resident)

// Workspace layout (float offsets)
constexpr size_t OFF_P0  = 0;                     // [NV][NG] precomputed W_ih0@E[v] + b_ih0 + b_hh0
constexpr size_t OFF_B1  = OFF_P0 + (size_t)NV * NG;   // [NG] b_ih1 + b_hh1
constexpr size_t OFF_H0  = OFF_B1 + NG;           // [2 sel][2 parity][NH]
constexpr size_t OFF_C0  = OFF_H0 + 4 * NH;
constexpr size_t OFF_H1  = OFF_C0 + 4 * NH;
constexpr size_t OFF_C1  = OFF_H1 + 4 * NH;
constexpr size_t OFF_TOK = OFF_C1 + 4 * NH;       // [NL] ints (token fed into step i+1)
constexpr size_t OFF_BAR = OFF_TOK + NL;          // [2] ints: arrive count, generation

#define ST(base, sel, par) ((base) + (((sel) * 2 + (par)) * NH))

typedef float v2f __attribute__((ext_vector_type(2)));
typedef float v8f __attribute__((ext_vector_type(8)));

// ---------------------------------------------------------------------------
// Init kernel: zero initial state, bias sums, static one-hot rows & tokens.
// (mask is a static input => every masked step's output row and next-token are
//  known ahead of time; write them once here instead of in the hot loop.)
// ---------------------------------------------------------------------------
__global__ void init_kernel(const int* __restrict__ song,
                            const unsigned char* __restrict__ mask,
                            const float* __restrict__ b_ih1,
                            const float* __restrict__ b_hh1,
                            float* __restrict__ ws,
                            float* __restrict__ out) {
  int tid = blockIdx.x * blockDim.x + threadIdx.x;
  int nthr = gridDim.x * blockDim.x;

  for (int u = tid; u < NH; u += nthr) {   // zero t=0 (parity 0, pass1) state
    (ws + OFF_H0)[u] = 0.f; (ws + OFF_C0)[u] = 0.f;
    (ws + OFF_H1)[u] = 0.f; (ws + OFF_C1)[u] = 0.f;
  }
  for (int g = tid; g < NG; g += nthr) (ws + OFF_B1)[g] = b_ih1[g] + b_hh1[g];

  int* tok = (int*)(ws + OFF_TOK);
  int* bar = (int*)(ws + OFF_BAR);
  if (tid == 0) { tok[0] = song[0]; bar[0] = 0; }  // bar[1] (gen) is relative-only

  for (int i = tid; i < NL; i += nthr) {
    if (i == 0 || mask[i]) {
      int s = song[i];
      float* row = out + (size_t)i * NV;
      for (int v = 0; v < NV; ++v) row[v] = (v == s) ? 1.f : 0.f;
      if (i > 0) tok[i] = s;               // teacher-forced next token
    }
  }
}

// ---------------------------------------------------------------------------
// Precompute P0[v][g] = (W_ih0 @ E[v])[g] + b_ih0[g] + b_hh0[g]   (fp32 GEMM,
// M=4096, N=97(pad 112), K=256) with V_WMMA_F32_16X16X4_F32, one wave/tile.
// A-frag (16x4): lane l<16 -> M=l, vgpr{0,1}=K{0,1}; lane l+16 -> K{2,3}.
// B-frag (4x16): lane l<16 -> N=l, vgpr{0,1}=K{0,1}; lane l+16 -> K{2,3}.
// C/D (16x16 f32): vgpr r -> M = 16*mt + r + 8*(lane>>4), N = 16*nt + (lane&15).
// Out-of-range vocab columns: B row pointer is clamped (finite garbage), the
// corresponding D columns are simply never stored -> branch-free K loop, EXEC
// stays all-ones around every WMMA.
// ---------------------------------------------------------------------------
__global__ void __launch_bounds__(256) precompute_p0(
    const float* __restrict__ E, const float* __restrict__ W_ih0,
    const float* __restrict__ b_ih0, const float* __restrict__ b_hh0,
    float* __restrict__ ws) {
  constexpr int NT = 7;                       // ceil(97/16) N tiles
  int wave = (blockIdx.x * blockDim.x + threadIdx.x) >> 5;
  int lane = threadIdx.x & 31;
  int mt = wave / NT, nt = wave % NT;         // 256 * 7 = 1792 waves exactly

  int mrow = mt * 16 + (lane & 15);
  int vcol = nt * 16 + (lane & 15);
  int khi  = (lane >> 4) * 2;

  const float* Arow = W_ih0 + (size_t)mrow * ND + khi;
  const float* Brow = E + (size_t)(vcol < NV ? vcol : (NV - 1)) * ND + khi;

  v8f acc = {0.f, 0.f, 0.f, 0.f, 0.f, 0.f, 0.f, 0.f};
#pragma unroll 16
  for (int k = 0; k < ND; k += 4) {
    v2f a, b;
    a[0] = Arow[k];
    a[1] = Arow[k + 1];
    b[0] = Brow[k];
    b[1] = Brow[k + 1];
    acc = __builtin_amdgcn_wmma_f32_16x16x4_f32(false, a, false, b,
                                                (short)0, acc, false, false);
  }
  if (vcol < NV) {
#pragma unroll
    for (int r = 0; r < 8; ++r) {
      int g = mt * 16 + r + 8 * (lane >> 4);
      (ws + OFF_P0)[(size_t)vcol * NG + g] = acc[r] + b_ih0[g] + b_hh0[g];
    }
  }
}

// ---------------------------------------------------------------------------
// Persistent LSTM scan kernel
// ---------------------------------------------------------------------------
__device__ __forceinline__ float sigf(float x) { return 1.f / (1.f + __expf(-x)); }

__device__ __forceinline__ void gbar(int* bar) {
  __threadfence();            // make this WG's state stores visible device-wide
  __syncthreads();
  if (threadIdx.x == 0) {
    int* cnt = bar;
    int* gen = bar + 1;
    int g = __hip_atomic_load(gen, __ATOMIC_ACQUIRE, __HIP_MEMORY_SCOPE_AGENT);
    if (__hip_atomic_fetch_add(cnt, 1, __ATOMIC_ACQ_REL, __HIP_MEMORY_SCOPE_AGENT) == NWG - 1) {
      __hip_atomic_store(cnt, 0, __ATOMIC_RELAXED, __HIP_MEMORY_SCOPE_AGENT);
      __hip_atomic_store(gen, g + 1, __ATOMIC_RELEASE, __HIP_MEMORY_SCOPE_AGENT);
    } else {
      while (__hip_atomic_load(gen, __ATOMIC_ACQUIRE, __HIP_MEMORY_SCOPE_AGENT) == g)
        __builtin_amdgcn_s_sleep(2);
    }
  }
  __syncthreads();
  __threadfence();            // acquire: discard stale cached lines
}

// Stage 1024 floats (4KB) global -> LDS with the CDNA5 async-LDS engine:
// GLOBAL_LOAD_ASYNC_TO_LDS_B128 (GV mode: VDST = per-lane LDS byte offset,
// VADDR = 64-bit global address). Generic pointers into LDS carry the LDS
// byte offset in addr[31:0] (aperture rule), so truncation yields VDST.
// Tracked by ASYNCcnt; caller must stage_wait() before __syncthreads().
__device__ __forceinline__ void stage1024(float* __restrict__ dst,
                                          const float* __restrict__ src) {
  unsigned lds_off = (unsigned)(unsigned long long)(dst + threadIdx.x * 4);
  unsigned long long gaddr =
      (unsigned long long)(const void*)(src + threadIdx.x * 4);
  asm volatile("global_load_async_to_lds_b128 %0, %1, off"
               :
               : "v"(lds_off), "v"(gaddr)
               : "memory");
}

__device__ __forceinline__ void stage_wait() {
  asm volatile("s_wait_asynccnt 0x0" ::: "memory");
}

__device__ __forceinline__ float dot512(const float* __restrict__ w,
                                        const float* __restrict__ v) {
  const float4* w4 = (const float4*)w;
  const float4* v4 = (const float4*)v;
  float acc = 0.f;
#pragma unroll 4
  for (int k = 0; k < 128; ++k) {
    float4 a = w4[k], b = v4[k];
    acc = fmaf(a.x, b.x, fmaf(a.y, b.y, fmaf(a.z, b.z, fmaf(a.w, b.w, acc))));
  }
  return acc;
}

__device__ __forceinline__ float dot1024(const float* __restrict__ w,
                                         const float* __restrict__ v) {
  const float4* w4 = (const float4*)w;
  const float4* v4 = (const float4*)v;
  float acc = 0.f;
#pragma unroll 4
  for (int k = 0; k < 256; ++k) {
    float4 a = w4[k], b = v4[k];
    acc = fmaf(a.x, b.x, fmaf(a.y, b.y, fmaf(a.z, b.z, fmaf(a.w, b.w, acc))));
  }
  return acc;
}

__global__ void __launch_bounds__(256, 1) lstm_persistent(
    const float* __restrict__ W_hh0, const float* __restrict__ W_ih1,
    const float* __restrict__ W_hh1, const float* __restrict__ Wfc,
    const float* __restrict__ bfc, const int* __restrict__ song,
    const unsigned char* __restrict__ mask, float* __restrict__ ws,
    float* __restrict__ out) {
  __shared__ __align__(16) float sX[NH];    // "x" vector stage
  __shared__ __align__(16) float sHv[NH];   // "h" vector stage
  __shared__ float sl[128];                 // FC logits for argmax

  float* h0b = ws + OFF_H0;  float* c0b = ws + OFF_C0;
  float* h1b = ws + OFF_H1;  float* c1b = ws + OFF_C1;
  const float* P0 = ws + OFF_P0;
  const float* B1 = ws + OFF_B1;
  int* tok = (int*)(ws + OFF_TOK);
  int* bar = (int*)(ws + OFF_BAR);

  const int gl = blockIdx.x * 256 + threadIdx.x;   // 0..8191
  const int lane = threadIdx.x & 31;

  for (int i = 1; i < NL; ++i) {
    const int par = i & 1, pp = (i - 1) & 1;
    const int m = mask[i];
    const int selp = (i == 1) ? 0 : (mask[i - 1] ? 1 : 0);

    // ---- Phase A: pass1 layer0 (all WGs, 8 lanes/unit, K split in halves) --
    {
      stage1024(sX, ST(h0b, selp, pp));
      stage_wait();
      __syncthreads();
      const int t = tok[i - 1];
      const int unit = gl >> 3, sub = gl & 7, gate = sub >> 1, half = sub & 1;
      const int row = gate * NH + unit;
      float acc = dot512(W_hh0 + (size_t)row * NH + half * 512, sX + half * 512);
      acc += __shfl_xor(acc, 1, 32);
      acc += P0[(size_t)t * NG + row];
      const int bl = lane & ~7;
      float gi = __shfl(acc, bl + 0, 32), gf = __shfl(acc, bl + 2, 32);
      float gg = __shfl(acc, bl + 4, 32), go = __shfl(acc, bl + 6, 32);
      if (sub == 0) {
        float cp = ST(c0b, selp, pp)[unit];
        float c2 = sigf(gf) * cp + sigf(gi) * tanhf(gg);
        ST(h0b, 0, par)[unit] = sigf(go) * tanhf(c2);
        ST(c0b, 0, par)[unit] = c2;
      }
    }
    gbar(bar);

    // ---- Phase B ----------------------------------------------------------
    if (!m) {  // pass1 layer1 on all 32 WGs (8 lanes/unit; half0=ih, half1=hh)
      stage1024(sX, ST(h0b, 0, par));
      stage1024(sHv, ST(h1b, selp, pp));
      stage_wait();
      __syncthreads();
      const int unit = gl >> 3, sub = gl & 7, gate = sub >> 1, half = sub & 1;
      const int row = gate * NH + unit;
      float acc = half ? dot1024(W_hh1 + (size_t)row * NH, sHv)
                       : dot1024(W_ih1 + (size_t)row * NH, sX);
      acc += __shfl_xor(acc, 1, 32);
      acc += B1[row];
      const int bl = lane & ~7;
      float gi = __shfl(acc, bl + 0, 32), gf = __shfl(acc, bl + 2, 32);
      float gg = __shfl(acc, bl + 4, 32), go = __shfl(acc, bl + 6, 32);
      if (sub == 0) {
        float cp = ST(c1b, selp, pp)[unit];
        float c2 = sigf(gf) * cp + sigf(gi) * tanhf(gg);
        ST(h1b, 0, par)[unit] = sigf(go) * tanhf(c2);
        ST(c1b, 0, par)[unit] = c2;
      }
    } else if (blockIdx.x < 16) {  // pass1 layer1 on 16 WGs (4 lanes/unit)
      stage1024(sX, ST(h0b, 0, par));
      stage1024(sHv, ST(h1b, selp, pp));
      stage_wait();
      __syncthreads();
      const int g2 = blockIdx.x * 256 + threadIdx.x;
      const int unit = g2 >> 2, gate = g2 & 3, row = gate * NH + unit;
      float acc = B1[row] + dot1024(W_ih1 + (size_t)row * NH, sX)
                          + dot1024(W_hh1 + (size_t)row * NH, sHv);
      const int bl = lane & ~3;
      float gi = __shfl(acc, bl + 0, 32), gf = __shfl(acc, bl + 1, 32);
      float gg = __shfl(acc, bl + 2, 32), go = __shfl(acc, bl + 3, 32);
      if (gate == 0) {
        float cp = ST(c1b, selp, pp)[unit];
        float c2 = sigf(gf) * cp + sigf(gi) * tanhf(gg);
        ST(h1b, 0, par)[unit] = sigf(go) * tanhf(c2);
        ST(c1b, 0, par)[unit] = c2;
      }
    } else {  // pass2 layer0 on 16 WGs (x = E[song[i]] via P0, h = h0 pass1)
      stage1024(sX, ST(h0b, 0, par));
      stage_wait();
      __syncthreads();
      const int g2 = (blockIdx.x - 16) * 256 + threadIdx.x;
      const int unit = g2 >> 2, gate = g2 & 3, row = gate * NH + unit;
      float acc = P0[(size_t)song[i] * NG + row] +
                  dot1024(W_hh0 + (size_t)row * NH, sX);
      const int bl = lane & ~3;
      float gi = __shfl(acc, bl + 0, 32), gf = __shfl(acc, bl + 1, 32);
      float gg = __shfl(acc, bl + 2, 32), go = __shfl(acc, bl + 3, 32);
      if (gate == 0) {
        float cp = ST(c0b, 0, par)[unit];
        float c2 = sigf(gf) * cp + sigf(gi) * tanhf(gg);
        ST(h0b, 1, par)[unit] = sigf(go) * tanhf(c2);
        ST(c0b, 1, par)[unit] = c2;
      }
    }
    gbar(bar);

    // ---- Phase C ----------------------------------------------------------
    if (m) {  // pass2 layer1 on all WGs (x = h0 pass2, h/c = h1/c1 pass1)
      stage1024(sX, ST(h0b, 1, par));
      stage1024(sHv, ST(h1b, 0, par));
      stage_wait();
      __syncthreads();
      const int unit = gl >> 3, sub = gl & 7, gate = sub >> 1, half = sub & 1;
      const int row = gate * NH + unit;
      float acc = half ? dot1024(W_hh1 + (size_t)row * NH, sHv)
                       : dot1024(W_ih1 + (size_t)row * NH, sX);
      acc += __shfl_xor(acc, 1, 32);
      acc += B1[row];
      const int bl = lane & ~7;
      float gi = __shfl(acc, bl + 0, 32), gf = __shfl(acc, bl + 2, 32);
      float gg = __shfl(acc, bl + 4, 32), go = __shfl(acc, bl + 6, 32);
      if (sub == 0) {
        float cp = ST(c1b, 0, par)[unit];
        float c2 = sigf(gf) * cp + sigf(gi) * tanhf(gg);
        ST(h1b, 1, par)[unit] = sigf(go) * tanhf(c2);
        ST(c1b, 1, par)[unit] = c2;
      }
    } else if (blockIdx.x == 0) {  // FC + argmax + emit logits, pick next token
      stage1024(sX, ST(h1b, 0, par));
      stage_wait();
      __syncthreads();
      const int tid = threadIdx.x, r = tid >> 1, hh = tid & 1;
      float acc = 0.f;
      if (r < NV) acc = dot512(Wfc + (size_t)r * NH + hh * 512, sX + hh * 512);
      acc += __shfl_xor(acc, 1, 32);
      if (r < NV && hh == 0) {
        acc += bfc[r];
        out[(size_t)i * NV + r] = acc;
        sl[r] = acc;
      }
      __syncthreads();
      if (tid == 0) {
        int best = 0; float bv = sl[0];
        for (int r2 = 1; r2 < NV; ++r2)
          if (sl[r2] > bv) { bv = sl[r2]; best = r2; }
        tok[i] = best;
      }
    }
    gbar(bar);
  }

  // ---- Epilogue: final hs/cs (order: out[L,V], hs_f[2,H], cs_f[2,H]) ------
  if (blockIdx.x == 0) {
    const int self = mask[NL - 1] ? 1 : 0, par = (NL - 1) & 1;
    float* fin = out + (size_t)NL * NV;
    for (int u = threadIdx.x; u < NH; u += 256) {
      fin[u]          = ST(h0b, self, par)[u];
      fin[NH + u]     = ST(h1b, self, par)[u];
      fin[2 * NH + u] = ST(c0b, self, par)[u];
      fin[3 * NH + u] = ST(c1b, self, par)[u];
    }
  }
}

// ---------------------------------------------------------------------------
extern "C" void kernel_launch(void* const* d_in, const int* in_sizes, int n_in,
                              void* d_out, int out_size, void* d_ws, size_t ws_size,
                              hipStream_t stream) {
  const float* E     = (const float*)d_in[0];
  const float* W_ih0 = (const float*)d_in[1];
  const float* W_hh0 = (const float*)d_in[2];
  const float* b_ih0 = (const float*)d_in[3];
  const float* b_hh0 = (const float*)d_in[4];
  const float* W_ih1 = (const float*)d_in[5];
  const float* W_hh1 = (const float*)d_in[6];
  const float* b_ih1 = (const float*)d_in[7];
  const float* b_hh1 = (const float*)d_in[8];
  const float* Wfc   = (const float*)d_in[9];
  const float* bfc   = (const float*)d_in[10];
  const int* song    = (const int*)d_in[11];
  const unsigned char* mask = (const unsigned char*)d_in[12];
  float* out = (float*)d_out;
  float* ws  = (float*)d_ws;

  init_kernel<<<dim3(8), dim3(256), 0, stream>>>(song, mask, b_ih1, b_hh1, ws, out);
  precompute_p0<<<dim3(224), dim3(256), 0, stream>>>(E, W_ih0, b_ih0, b_hh0, ws);
  lstm_persistent<<<dim3(NWG), dim3(256), 0, stream>>>(W_hh0, W_ih1, W_hh1, Wfc,
                                                       bfc, song, mask, ws, out);
}